// GATModel_13477607375483
// MI455X (gfx1250) — compile-verified
//
#include <hip/hip_runtime.h>
#include <hip/hip_bf16.h>
#include <math.h>

#define NEG_SLOPE 0.2f

typedef __attribute__((ext_vector_type(2))) float v2f;
typedef __attribute__((ext_vector_type(8))) float v8f;

// ---------------------------------------------------------------------------
// Dense transform H = X @ W using V_WMMA_F32_16X16X4_F32 (exact fp32 path).
// One wave32 computes a 16 x (16*NT) strip; one A fragment feeds NT WMMAs per
// K-step. Software-pipelined: fragments for step k+4 are loaded before the
// WMMAs of step k are issued, so the per-WMMA s_wait_loadcnt covers loads that
// have been in flight for a full iteration.
// A layout: lanes 0-15 rows M, VGPR0/1 = K {0,1}; lanes 16-31 = K {2,3}.
// C/D layout: VGPR r -> rows r (lanes 0-15) / r+8 (lanes 16-31).
// ---------------------------------------------------------------------------
template <int NT>
__global__ __launch_bounds__(32)
void gemm_wmma_f32(const float* __restrict__ X, const float* __restrict__ W,
                   float* __restrict__ Hout, int K, int HD, int HDW) {
  const int col0 = blockIdx.x * (16 * NT);
  const int row0 = blockIdx.y * 16;
  const int lane = threadIdx.x & 31;
  const int half = lane >> 4;     // which K pair / which row half
  const int l    = lane & 15;

  v8f zero = {};
  v8f acc[NT];
#pragma unroll
  for (int t = 0; t < NT; ++t) acc[t] = zero;

  const float* xr = X + (size_t)(row0 + l) * K + 2 * half;
  const float* wb = W + (size_t)(2 * half) * HDW + col0 + l;

  // prologue: load K-step 0
  v2f a;
  a.x = xr[0];
  a.y = xr[1];
  v2f b[NT];
#pragma unroll
  for (int t = 0; t < NT; ++t) {
    b[t].x = wb[t * 16];
    b[t].y = wb[t * 16 + HDW];
  }

  // steady state: load step k0+4, compute step k0
  for (int k0 = 0; k0 + 4 < K; k0 += 4) {
    v2f an;
    an.x = xr[k0 + 4];
    an.y = xr[k0 + 5];
    const float* wr = wb + (size_t)(k0 + 4) * HDW;
    v2f bn[NT];
#pragma unroll
    for (int t = 0; t < NT; ++t) {
      bn[t].x = wr[t * 16];
      bn[t].y = wr[t * 16 + HDW];
    }
#pragma unroll
    for (int t = 0; t < NT; ++t)
      acc[t] = __builtin_amdgcn_wmma_f32_16x16x4_f32(
          false, a, false, b[t], (short)0, acc[t], false, false);
    a = an;
#pragma unroll
    for (int t = 0; t < NT; ++t) b[t] = bn[t];
  }
  // epilogue: last K-step
#pragma unroll
  for (int t = 0; t < NT; ++t)
    acc[t] = __builtin_amdgcn_wmma_f32_16x16x4_f32(
        false, a, false, b[t], (short)0, acc[t], false, false);

  const int orow = row0 + 8 * half;
#pragma unroll
  for (int t = 0; t < NT; ++t) {
    const int col = col0 + t * 16 + l;
    if (col < HD) {
#pragma unroll
      for (int r = 0; r < 8; ++r)
        Hout[(size_t)(orow + r) * HD + col] = acc[t][r];
    }
  }
}

// zero-pad W [K x HD] into Wp [K x HDW]
__global__ void pad_w(const float* __restrict__ W, float* __restrict__ Wp,
                      int K, int HD, int HDW) {
  const int i = blockIdx.x * blockDim.x + threadIdx.x;
  if (i >= K * HDW) return;
  const int r = i / HDW, c = i - r * HDW;
  Wp[i] = (c < HD) ? W[r * HD + c] : 0.0f;
}

// ---------------------------------------------------------------------------
// e_src[n,h] = dot(h[n,h,:], a_src[h,:]); e_dst likewise. One wave per (n,h).
// ---------------------------------------------------------------------------
__global__ void attn_coef(const float* __restrict__ Hf,
                          const float* __restrict__ asrc,
                          const float* __restrict__ adst,
                          float* __restrict__ es, float* __restrict__ ed,
                          int Nn, int Hh, int D) {
  const int wid  = (blockIdx.x * blockDim.x + threadIdx.x) >> 5;
  const int lane = threadIdx.x & 31;
  if (wid >= Nn * Hh) return;
  const int n = wid / Hh;
  const int h = wid - n * Hh;
  const float* hp = Hf + (size_t)n * Hh * D + (size_t)h * D;
  const float* as = asrc + h * D;
  const float* ad = adst + h * D;
  float s = 0.0f, d = 0.0f;
  for (int i = lane; i < D; i += 32) {
    const float v = hp[i];
    s += v * as[i];
    d += v * ad[i];
  }
#pragma unroll
  for (int off = 16; off > 0; off >>= 1) {
    s += __shfl_xor(s, off, 32);
    d += __shfl_xor(d, off, 32);
  }
  if (lane == 0) { es[wid] = s; ed[wid] = d; }
}

__global__ void fill_f32(float* __restrict__ p, float v, int n) {
  const int i = blockIdx.x * blockDim.x + threadIdx.x;
  if (i < n) p[i] = v;
}

// float atomic max via signed/unsigned integer punning (init value is -inf)
__device__ inline void atomic_max_f32(float* addr, float val) {
  if (val >= 0.0f)
    atomicMax((int*)addr, __float_as_int(val));
  else
    atomicMin((unsigned int*)addr, __float_as_uint(val));
}

__device__ inline void edge_endpoints(const int* __restrict__ ei, int E, int e,
                                      int& src, int& dst) {
  if (e < E) { src = ei[e]; dst = ei[E + e]; }
  else       { src = e - E; dst = e - E; }     // appended self loops
}

// pass 1: segment max of leaky_relu(e_src[src]+e_dst[dst]) over dst
__global__ void edge_max(const int* __restrict__ ei, int E, int Etot,
                         const float* __restrict__ es, const float* __restrict__ ed,
                         float* __restrict__ mseg, int Hh) {
  const int idx = blockIdx.x * blockDim.x + threadIdx.x;
  if (idx >= Etot * Hh) return;
  const int e = idx / Hh;
  const int h = idx - e * Hh;
  int src, dst; edge_endpoints(ei, E, e, src, dst);
  float v = es[src * Hh + h] + ed[dst * Hh + h];
  v = (v >= 0.0f) ? v : v * NEG_SLOPE;
  atomic_max_f32(&mseg[dst * Hh + h], v);
}

// pass 2: ex = exp(e - m[dst]); denom[dst] += ex; stash ex per edge
__global__ void edge_expsum(const int* __restrict__ ei, int E, int Etot,
                            const float* __restrict__ es, const float* __restrict__ ed,
                            const float* __restrict__ mseg, float* __restrict__ dseg,
                            float* __restrict__ exb, int Hh) {
  const int idx = blockIdx.x * blockDim.x + threadIdx.x;
  if (idx >= Etot * Hh) return;
  const int e = idx / Hh;
  const int h = idx - e * Hh;
  int src, dst; edge_endpoints(ei, E, e, src, dst);
  float v = es[src * Hh + h] + ed[dst * Hh + h];
  v = (v >= 0.0f) ? v : v * NEG_SLOPE;
  const float ex = expf(v - mseg[dst * Hh + h]);
  exb[idx] = ex;
  atomicAdd(&dseg[dst * Hh + h], ex);
}

// pass 3: out[dst,h,:] += h[src,h,:] * alpha. One wave per (edge, head).
__global__ void edge_aggr(const int* __restrict__ ei, int E, int Etot,
                          const float* __restrict__ exb, const float* __restrict__ dseg,
                          const float* __restrict__ Hf, float* __restrict__ out,
                          int Hh, int D) {
  const int wid  = (blockIdx.x * blockDim.x + threadIdx.x) >> 5;
  const int lane = threadIdx.x & 31;
  if (wid >= Etot * Hh) return;
  const int e = wid / Hh;
  const int h = wid - e * Hh;
  int src, dst; edge_endpoints(ei, E, e, src, dst);
  const float alpha = exb[wid] / (dseg[dst * Hh + h] + 1e-16f);
  const float* hs = Hf  + (size_t)src * Hh * D + (size_t)h * D;
  float*       op = out + (size_t)dst * Hh * D + (size_t)h * D;
  for (int i = lane; i < D; i += 32)
    atomicAdd(&op[i], hs[i] * alpha);
}

// bias + ELU; hdMask = HD-1 (HD is a power of two for the ELU layers)
__global__ void bias_elu(const float* __restrict__ acc, const float* __restrict__ bias,
                         float* __restrict__ out, int total, int hdMask) {
  const int i = blockIdx.x * blockDim.x + threadIdx.x;
  if (i >= total) return;
  const float v = acc[i] + bias[i & hdMask];
  out[i] = (v > 0.0f) ? v : expm1f(v);
}

// bias + numerically stable log_softmax over C=40 classes; one wave per node.
__global__ void bias_logsoftmax(const float* __restrict__ acc,
                                const float* __restrict__ bias,
                                float* __restrict__ out, int Nn, int C) {
  const int wid  = (blockIdx.x * blockDim.x + threadIdx.x) >> 5;
  const int lane = threadIdx.x & 31;
  if (wid >= Nn) return;
  const float* a = acc + (size_t)wid * C;
  const float v1 = (lane      < C) ? a[lane]      + bias[lane]      : -INFINITY;
  const float v2 = (lane + 32 < C) ? a[lane + 32] + bias[lane + 32] : -INFINITY;
  float mx = fmaxf(v1, v2);
#pragma unroll
  for (int off = 16; off > 0; off >>= 1) mx = fmaxf(mx, __shfl_xor(mx, off, 32));
  float s = ((lane < C) ? expf(v1 - mx) : 0.0f) +
            ((lane + 32 < C) ? expf(v2 - mx) : 0.0f);
#pragma unroll
  for (int off = 16; off > 0; off >>= 1) s += __shfl_xor(s, off, 32);
  const float lse = mx + logf(s);
  if (lane < C)      out[(size_t)wid * C + lane]      = v1 - lse;
  if (lane + 32 < C) out[(size_t)wid * C + lane + 32] = v2 - lse;
}

extern "C" void kernel_launch(void* const* d_in, const int* in_sizes, int n_in,
                              void* d_out, int out_size, void* d_ws, size_t ws_size,
                              hipStream_t stream) {
  const float* x   = (const float*)d_in[0];
  const int*   ei  = (const int*)  d_in[1];
  const float* W1  = (const float*)d_in[2];
  const float* a1s = (const float*)d_in[3];
  const float* a1d = (const float*)d_in[4];
  const float* b1  = (const float*)d_in[5];
  const float* W2  = (const float*)d_in[6];
  const float* a2s = (const float*)d_in[7];
  const float* a2d = (const float*)d_in[8];
  const float* b2  = (const float*)d_in[9];
  const float* W3  = (const float*)d_in[10];
  const float* a3s = (const float*)d_in[11];
  const float* a3d = (const float*)d_in[12];
  const float* b3  = (const float*)d_in[13];
  float* out = (float*)d_out;

  const int N    = in_sizes[0] / 128;   // 20000
  const int E    = in_sizes[1] / 2;     // 320000
  const int Etot = E + N;               // + self loops

  // ---- workspace carve ----
  char* ws = (char*)d_ws;
  size_t off = 0;
  auto alloc = [&](size_t nfloats) -> float* {
    float* p = (float*)(ws + off);
    off += (nfloats * sizeof(float) + 255) & ~(size_t)255;
    return p;
  };
  float* bufH = alloc((size_t)N * 256);   // transformed features (current layer)
  float* bufO = alloc((size_t)N * 256);   // aggregation accumulator / next input
  float* es   = alloc((size_t)N * 4);
  float* ed   = alloc((size_t)N * 4);
  float* mseg = alloc((size_t)N * 4);
  float* dseg = alloc((size_t)N * 4);
  float* exb  = alloc((size_t)Etot * 4);
  float* w3p  = alloc((size_t)128 * 48);  // layer-3 weights padded 40 -> 48 cols
  (void)ws_size; (void)n_in; (void)out_size;

  const float NEG_INF = -INFINITY;

  // run softmax-attention aggregation for one layer (after its GEMM)
  auto run_attn = [&](const float* asrc, const float* adst, int Hh, int D) {
    const int HD = Hh * D;
    const int nwaves = N * Hh;
    attn_coef<<<(nwaves * 32 + 255) / 256, 256, 0, stream>>>(bufH, asrc, adst,
                                                             es, ed, N, Hh, D);
    fill_f32<<<(N * Hh + 255) / 256, 256, 0, stream>>>(mseg, NEG_INF, N * Hh);
    fill_f32<<<(N * Hh + 255) / 256, 256, 0, stream>>>(dseg, 0.0f, N * Hh);
    fill_f32<<<(N * HD + 255) / 256, 256, 0, stream>>>(bufO, 0.0f, N * HD);

    const int eh = Etot * Hh;
    edge_max<<<(eh + 255) / 256, 256, 0, stream>>>(ei, E, Etot, es, ed, mseg, Hh);
    edge_expsum<<<(eh + 255) / 256, 256, 0, stream>>>(ei, E, Etot, es, ed,
                                                      mseg, dseg, exb, Hh);
    const long long ethreads = (long long)eh * 32;
    edge_aggr<<<(unsigned)((ethreads + 255) / 256), 256, 0, stream>>>(
        ei, E, Etot, exb, dseg, bufH, bufO, Hh, D);
  };

  // ---- layer 1: 128 -> 4 x 64, ELU ----
  {
    dim3 g(256 / 64, N / 16);
    gemm_wmma_f32<4><<<g, 32, 0, stream>>>(x, W1, bufH, 128, 256, 256);
    run_attn(a1s, a1d, 4, 64);
    bias_elu<<<(N * 256 + 255) / 256, 256, 0, stream>>>(bufO, b1, bufO, N * 256, 255);
  }
  // ---- layer 2: 256 -> 2 x 64, ELU (bufO read by GEMM, then reused as accum) ----
  {
    dim3 g(128 / 64, N / 16);
    gemm_wmma_f32<4><<<g, 32, 0, stream>>>(bufO, W2, bufH, 256, 128, 128);
    run_attn(a2s, a2d, 2, 64);
    bias_elu<<<(N * 128 + 255) / 256, 256, 0, stream>>>(bufO, b2, bufO, N * 128, 127);
  }
  // ---- layer 3: 128 -> 1 x 40, log_softmax ----
  {
    pad_w<<<(128 * 48 + 255) / 256, 256, 0, stream>>>(W3, w3p, 128, 40, 48);
    dim3 g(1, N / 16);
    gemm_wmma_f32<3><<<g, 32, 0, stream>>>(bufO, w3p, bufH, 128, 40, 48);
    run_attn(a3s, a3d, 1, 40);
    bias_logsoftmax<<<(N * 32 + 255) / 256, 256, 0, stream>>>(bufO, b3, out, N, 40);
  }
}